// FM_Layer_2353642078691
// MI455X (gfx1250) — compile-verified
//
#include <hip/hip_runtime.h>

#define NUM_DENSE 13
#define NUM_SPARSE 26
#define FEAT_NUM 40000
#define FEATURE_LENGTH (NUM_SPARSE * FEAT_NUM + NUM_DENSE)  // 1040013
#define KDIM 64
#define BATCH 4096
#define ROWS_PER_BLOCK 16

typedef __attribute__((ext_vector_type(2))) float v2f;
typedef __attribute__((ext_vector_type(8))) float v8f;

// One block = 16 batch rows, 128 threads = 4 waves (wave32).
// Wave w owns k-columns [16w, 16w+16). Its WMMA C tile (v8f) holds
// sv[m][k] for m = v + 8*(lane>=16), k = 16w + (lane&15) -- the exact
// V_WMMA 32-bit C/D layout -- so sparse gathers accumulate straight into it.
__global__ __launch_bounds__(128) void fm_layer_kernel(
    const float* __restrict__ dense,    // [BATCH][NUM_DENSE]
    const int*   __restrict__ sparse,   // [BATCH][NUM_SPARSE]
    const float* __restrict__ w0,       // [1]
    const float* __restrict__ w,        // [FEATURE_LENGTH]
    const float* __restrict__ V,        // [KDIM][FEATURE_LENGTH] row-major
    float* __restrict__ out)            // [BATCH]
{
    __shared__ int   sfeat[ROWS_PER_BLOCK * NUM_SPARSE];  // absolute feature ids
    __shared__ float ssec[ROWS_PER_BLOCK];                // per-row second-order sum

    const int tid  = threadIdx.x;
    const int wave = tid >> 5;       // 0..3
    const int lane = tid & 31;
    const int lo   = lane & 15;      // N index within tile
    const int hi   = lane >> 4;      // half-wave select
    const int base = blockIdx.x * ROWS_PER_BLOCK;

    // Stage sparse feature indices (idx = raw + 13 + 40000*s) into LDS.
    for (int i = tid; i < ROWS_PER_BLOCK * NUM_SPARSE; i += 128) {
        int m = i / NUM_SPARSE, s = i - m * NUM_SPARSE;
        sfeat[i] = sparse[(base + m) * NUM_SPARSE + s] + NUM_DENSE + s * FEAT_NUM;
    }
    if (tid < ROWS_PER_BLOCK) ssec[tid] = 0.0f;
    __syncthreads();

    const int    k    = wave * 16 + lo;                 // this lane's k column
    const size_t vrow = (size_t)k * FEATURE_LENGTH;     // &V[k][0] offset

    // ---- Dense part via exact fp32 WMMA (K=13 zero-padded to 16, 4 chunks of 4):
    //      c_sv += dense    @ VtDense     ;  c_s2 += dense^2 @ VtDense^2
    v8f c_sv = {0.f,0.f,0.f,0.f,0.f,0.f,0.f,0.f};
    v8f c_s2 = {0.f,0.f,0.f,0.f,0.f,0.f,0.f,0.f};
#pragma unroll
    for (int c = 0; c < 4; ++c) {
        v2f a, b, a2, b2;
#pragma unroll
        for (int v = 0; v < 2; ++v) {
            // A 16x4 layout: VGPR v, lane: M = lo, Kidx = v + 2*hi
            // B 4x16 layout: VGPR v, lane: Kidx = v + 2*hi, N = lo
            int d = 4 * c + v + 2 * hi;
            float av = (d < NUM_DENSE) ? dense[(base + lo) * NUM_DENSE + d] : 0.0f;
            float bv = (d < NUM_DENSE) ? V[vrow + d] : 0.0f;
            a[v] = av;  b[v] = bv;  a2[v] = av * av;  b2[v] = bv * bv;
        }
        c_sv = __builtin_amdgcn_wmma_f32_16x16x4_f32(
                   false, a,  false, b,  (short)0, c_sv, false, false);
        c_s2 = __builtin_amdgcn_wmma_f32_16x16x4_f32(
                   false, a2, false, b2, (short)0, c_s2, false, false);
    }

    // ---- Sparse gathers accumulate into the WMMA-layout accumulators.
    for (int s = 0; s < NUM_SPARSE; ++s) {
#pragma unroll
        for (int v = 0; v < 8; ++v) {
            int   m   = v + 8 * hi;                       // batch row in tile
            int   f   = sfeat[m * NUM_SPARSE + s];
            float val = V[vrow + (size_t)f];
            c_sv[v] += val;
            c_s2[v] += val * val;
        }
    }

    // ---- second[m] = 0.5 * sum_k (sv^2 - s2v2): reduce over this wave's 16 k,
    //      then LDS-atomic combine across the 4 waves.
    v8f part;
#pragma unroll
    for (int v = 0; v < 8; ++v) part[v] = c_sv[v] * c_sv[v] - c_s2[v];
#pragma unroll
    for (int msk = 1; msk <= 8; msk <<= 1) {
#pragma unroll
        for (int v = 0; v < 8; ++v)
            part[v] += __shfl_xor(part[v], msk, 32);      // stays within 16-lane half
    }
    if (lo == 0) {
#pragma unroll
        for (int v = 0; v < 8; ++v)
            atomicAdd(&ssec[v + 8 * hi], part[v]);        // ds_add_f32
    }
    __syncthreads();

    // ---- First-order term + final output, one thread per row.
    if (tid < ROWS_PER_BLOCK) {
        int m = tid;
        float fs = w0[0];
#pragma unroll
        for (int d = 0; d < NUM_DENSE; ++d)
            fs += dense[(base + m) * NUM_DENSE + d] * w[d];
#pragma unroll
        for (int s = 0; s < NUM_SPARSE; ++s)
            fs += w[sfeat[m * NUM_SPARSE + s]];
        out[base + m] = fs + 0.5f * ssec[m];
    }
}

extern "C" void kernel_launch(void* const* d_in, const int* in_sizes, int n_in,
                              void* d_out, int out_size, void* d_ws, size_t ws_size,
                              hipStream_t stream) {
    const float* dense  = (const float*)d_in[0];
    const int*   sparse = (const int*)  d_in[1];
    const float* w0     = (const float*)d_in[2];
    const float* w      = (const float*)d_in[3];
    const float* V      = (const float*)d_in[4];
    float*       out    = (float*)d_out;

    dim3 grid(BATCH / ROWS_PER_BLOCK);   // 256 blocks
    dim3 block(128);                     // 4 waves of 32
    hipLaunchKernelGGL(fm_layer_kernel, grid, block, 0, stream,
                       dense, sparse, w0, w, V, out);
}